// TopicAttention_8538394984662
// MI455X (gfx1250) — compile-verified
//
#include <hip/hip_runtime.h>

typedef float v2f __attribute__((ext_vector_type(2)));
typedef float v8f __attribute__((ext_vector_type(8)));

#define B_   512
#define T_   500
#define ENC_ 1024
#define DEC_ 1024

// ---------------------------------------------------------------------------
// Kernel 1: sub[b,e] = sum_d dec[b,d] * W[e,d]   (M=512, N=1024, K=1024, f32)
// One wave per 16x16 tile, V_WMMA_F32_16X16X4_F32 over the K loop.
// A fragment (16x4, MxK): lane<16 -> M=lane, {V0,V1}={K0,K1};
//                         lane>=16 -> M=lane-16, {V0,V1}={K2,K3}.
// B fragment (4x16, KxN): lane<16 -> N=lane, {V0,V1}={K0,K1};
//                         lane>=16 -> N=lane-16, {V0,V1}={K2,K3}.
// D (16x16 f32): VGPR i, lane<16 -> (M=i, N=lane); lane>=16 -> (M=i+8, N=lane-16).
// ---------------------------------------------------------------------------
__global__ __launch_bounds__(32) void sub_gemm_wmma(const float* __restrict__ dec,
                                                    const float* __restrict__ W,
                                                    float* __restrict__ sub) {
    const int e0   = blockIdx.x * 16;
    const int b0   = blockIdx.y * 16;
    const int lane = threadIdx.x;
    const int half = lane >> 4;     // 0: K pair {0,1}, 1: K pair {2,3}
    const int lm   = lane & 15;

    const float* arow = dec + (size_t)(b0 + lm) * DEC_ + 2 * half;  // row of A = dec row b0+lm
    const float* brow = W   + (size_t)(e0 + lm) * DEC_ + 2 * half;  // col of B = W row e0+lm

    v8f c = {};
#pragma unroll 8
    for (int k = 0; k < DEC_; k += 4) {
        v2f a = *(const v2f*)(arow + k);   // 8B aligned (even float offset)
        v2f b = *(const v2f*)(brow + k);
        c = __builtin_amdgcn_wmma_f32_16x16x4_f32(
                /*neg_a=*/false, a, /*neg_b=*/false, b,
                /*c_mod=*/(short)0, c, /*reuse_a=*/false, /*reuse_b=*/false);
    }

#pragma unroll
    for (int i = 0; i < 8; ++i) {
        const int m = i + 8 * half;
        sub[(size_t)(b0 + m) * ENC_ + (e0 + lm)] = c[i];
    }
}

// ---------------------------------------------------------------------------
// Kernel 2: fused masked-softmax attention, single streaming pass (online
// softmax).  One 256-thread block (8 wave32) per batch row; each thread owns
// float4 slice e = 4*tid .. 4*tid+3 of the 1024-wide feature.  Rows with
// t >= len contribute exactly 0 and are never read.
// ---------------------------------------------------------------------------
__global__ __launch_bounds__(256) void attn_fused(const float* __restrict__ enc,
                                                  const int*   __restrict__ lens,
                                                  const float* __restrict__ sub,
                                                  float* __restrict__ attn_out,
                                                  float* __restrict__ sum_out) {
    __shared__ float s_scores[T_];
    __shared__ float wred[2][8];   // double-buffered cross-wave reduce slots

    const int b    = blockIdx.x;
    const int tid  = threadIdx.x;
    const int lane = tid & 31;
    const int wave = tid >> 5;
    const int len  = lens[b];      // 1..T

    const float4 sv = ((const float4*)(sub + (size_t)b * ENC_))[tid];

    const float4* encb = (const float4*)(enc + (size_t)b * T_ * ENC_);

    float4 acc = make_float4(0.f, 0.f, 0.f, 0.f);
    float  m   = -__builtin_inff();
    float  l   = 0.f;

    for (int t = 0; t < len; ++t) {
        const float4 ev = encb[(size_t)t * (ENC_ / 4) + tid];
        if (t + 1 < len)
            __builtin_prefetch(&encb[(size_t)(t + 1) * (ENC_ / 4) + tid], 0, 1);

        // per-thread partial dot over 4 features
        float p = sv.x * ev.x + sv.y * ev.y + sv.z * ev.z + sv.w * ev.w;

        // wave32 reduce
#pragma unroll
        for (int off = 16; off >= 1; off >>= 1)
            p += __shfl_xor(p, off, 32);
        if (lane == 0) wred[t & 1][wave] = p;
        __syncthreads();

        const float* wr = wred[t & 1];
        const float score = wr[0] + wr[1] + wr[2] + wr[3] +
                            wr[4] + wr[5] + wr[6] + wr[7];
        if (tid == 0) s_scores[t] = score;

        // online softmax update (identical scalar work in every thread)
        const float mn   = fmaxf(m, score);
        const float corr = __expf(m - mn);       // first iter: exp(-inf)=0
        const float pe   = __expf(score - mn);
        l = l * corr + pe;
        acc.x = acc.x * corr + pe * ev.x;
        acc.y = acc.y * corr + pe * ev.y;
        acc.z = acc.z * corr + pe * ev.z;
        acc.w = acc.w * corr + pe * ev.w;
        m = mn;
    }
    __syncthreads();   // make s_scores visible

    const float inv = 1.0f / l;
    float4 r;
    r.x = acc.x * inv; r.y = acc.y * inv; r.z = acc.z * inv; r.w = acc.w * inv;
    ((float4*)(sum_out + (size_t)b * ENC_))[tid] = r;

    for (int t = tid; t < T_; t += 256) {
        attn_out[(size_t)b * T_ + t] =
            (t < len) ? __expf(s_scores[t] - m) * inv : 0.0f;
    }
}

// ---------------------------------------------------------------------------
extern "C" void kernel_launch(void* const* d_in, const int* in_sizes, int n_in,
                              void* d_out, int out_size, void* d_ws, size_t ws_size,
                              hipStream_t stream) {
    const float* dec  = (const float*)d_in[0];   // [B, DEC]
    const float* enc  = (const float*)d_in[1];   // [B, T, ENC]
    const int*   lens = (const int*)  d_in[2];   // [B]
    const float* W    = (const float*)d_in[3];   // [ENC, DEC]

    float* out      = (float*)d_out;
    float* attn_out = out;                        // B*T floats
    float* sum_out  = out + (size_t)B_ * T_;      // B*ENC floats

    float* sub = (float*)d_ws;                    // B*ENC floats = 2 MB scratch

    dim3 g1(ENC_ / 16, B_ / 16);                  // 64 x 32 tiles
    sub_gemm_wmma<<<g1, 32, 0, stream>>>(dec, W, sub);

    attn_fused<<<B_, 256, 0, stream>>>(enc, lens, sub, attn_out, sum_out);
}